// BinaryTreeLVeG_48249662603454
// MI455X (gfx1250) — compile-verified
//
#include <hip/hip_runtime.h>
#include <math.h>

// ---------------------------------------------------------------------------
// Problem constants (from reference)
// ---------------------------------------------------------------------------
#define LL      5
#define CC      4
#define D_IN    300
#define NNODES  255
#define BATCH   128
#define NCOL    20          // L*C = L*C*G = 20
#define NROWS   (BATCH * NNODES)   // 32640
#define LOG2PI_F 1.8378770664093453f
#define NEG_INF  (-3.0e38f)

typedef float v2f __attribute__((ext_vector_type(2)));
typedef float v8f __attribute__((ext_vector_type(8)));

// ---------------------------------------------------------------------------
// Kernel 1: projection GEMM via V_WMMA_F32_16X16X4_F32
//   out = hidden(32640x300) @ [W_w | W_mu | W_var](300x60)  (+bias, svar clip)
// One block = 16 output rows. A tile staged in LDS. 4 waves, one 16-col tile
// each (cols 0..63, valid < 60). K loop: 75 steps of 4.
//
// Padded columns 60..63 load a safe finite value (Wvr col 0) WITHOUT masking:
// in D = A x B each output column depends only on its own B column, and those
// output columns are never stored, so no mask multiply is needed. This keeps
// the hot loop at: 1 ds_load_b64 + 2 global_load_b32 + 1 wmma per K-step,
// with EXEC all-ones at every WMMA (ISA requirement).
// ---------------------------------------------------------------------------
__global__ __launch_bounds__(128) void lveg_proj_gemm(
    const float* __restrict__ hidden,
    const float* __restrict__ Ww,  const float* __restrict__ bw,
    const float* __restrict__ Wmu, const float* __restrict__ bmu,
    const float* __restrict__ Wvr, const float* __restrict__ bvr,
    float* __restrict__ sw, float* __restrict__ smu, float* __restrict__ svar)
{
    __shared__ float lA[16 * D_IN];

    const int tid     = threadIdx.x;
    const int rowBase = blockIdx.x * 16;

    // Stage 16 x 300 A-tile into LDS (coalesced)
    for (int idx = tid; idx < 16 * D_IN; idx += 128) {
        int r = idx / D_IN;
        int c = idx - r * D_IN;
        lA[idx] = hidden[(size_t)(rowBase + r) * D_IN + c];
    }
    __syncthreads();

    const int wave = tid >> 5;
    const int lane = tid & 31;
    const int half = lane >> 4;     // 0: K0/K1 rows, 1: K2/K3 rows
    const int l16  = lane & 15;
    const int col  = wave * 16 + l16;   // logical concat column 0..63

    // Per-lane weight matrix selection; invalid lanes (col>=60) use a safe
    // pointer -- their results are discarded at store time.
    const float* W    = Wvr;
    const float* bias = bvr;
    int wc = 0;
    if (col < 20)      { W = Ww;  bias = bw;  wc = col;      }
    else if (col < 40) { W = Wmu; bias = bmu; wc = col - 20; }
    else if (col < 60) { W = Wvr; bias = bvr; wc = col - 40; }

    const int kk = half * 2;
    v8f acc = {};
    for (int k = 0; k < D_IN; k += 4) {
        v2f a, b;
        a.x = lA[l16 * D_IN + k + kk];
        a.y = lA[l16 * D_IN + k + kk + 1];
        b.x = W[(size_t)(k + kk)     * NCOL + wc];
        b.y = W[(size_t)(k + kk + 1) * NCOL + wc];
        acc = __builtin_amdgcn_wmma_f32_16x16x4_f32(
                  /*neg_a=*/false, a, /*neg_b=*/false, b,
                  /*c_mod=*/(short)0, acc, /*reuse_a=*/false, /*reuse_b=*/false);
    }

    const float bv = bias[wc];
    // D layout: VGPR r -> M = r + 8*half (lanes 0-15 / 16-31), N = lane&15
#pragma unroll
    for (int r = 0; r < 8; ++r) {
        const int    row = rowBase + r + half * 8;
        const float  v   = acc[r] + bv;
        if (col < 20) {
            sw[(size_t)row * NCOL + col] = v;
        } else if (col < 40) {
            smu[(size_t)row * NCOL + (col - 20)] = v;
        } else if (col < 60) {
            float vc = fminf(10.0f, fmaxf(-10.0f, v));
            svar[(size_t)row * NCOL + (col - 40)] = vc;
        }
    }
}

// ---------------------------------------------------------------------------
// Kernel 2: one tree level. One thread per (b, parent m, parent-label i).
// Computes the 5x5x4x4 score grid, stable top-4, writes (w, mu, var).
// ---------------------------------------------------------------------------
__global__ void lveg_tree_level(
    const float* __restrict__ cw, const float* __restrict__ cmu,
    const float* __restrict__ cvar,
    int childRowStride, int childBase,
    const float* __restrict__ sw, const float* __restrict__ smu,
    const float* __restrict__ svar,
    int start, int M,
    const float* __restrict__ t_w,
    const float* __restrict__ t_mu_lc, const float* __restrict__ t_var_lc,
    const float* __restrict__ t_mu_rc, const float* __restrict__ t_var_rc,
    const float* __restrict__ t_mu_p,  const float* __restrict__ t_var_p,
    float* __restrict__ ow, float* __restrict__ omu, float* __restrict__ ovar)
{
    const int t     = blockIdx.x * blockDim.x + threadIdx.x;
    const int total = BATCH * M * LL;
    if (t >= total) return;

    const int i = t % LL;
    const int m = (t / LL) % M;
    const int b = t / (LL * M);

    const size_t lOff = ((size_t)b * childRowStride + childBase + 2 * m) * NCOL;
    const size_t rOff = lOff + NCOL;

    // Parent sufficient statistics, row i
    const size_t pOff = ((size_t)b * NNODES + start + m) * NCOL + i * CC;
    float pw[CC], pmu[CC], pvr[CC], v1p[CC];
#pragma unroll
    for (int cp = 0; cp < CC; ++cp) {
        pw[cp]  = sw[pOff + cp];
        pmu[cp] = smu[pOff + cp];
        float v = svar[pOff + cp];
        pvr[cp] = v;
        v1p[cp] = __expf(2.0f * v);
    }

    float best_v[CC] = {NEG_INF, NEG_INF, NEG_INF, NEG_INF};
    int   best_i[CC] = {0, 0, 0, 0};

    for (int j = 0; j < LL; ++j) {
        // Left child, label row j
        float ljw[CC], ljmu[CC], v1l[CC];
#pragma unroll
        for (int cl = 0; cl < CC; ++cl) {
            const size_t o = lOff + j * CC + cl;
            ljw[cl]  = cw[o];
            ljmu[cl] = cmu[o];
            v1l[cl]  = __expf(2.0f * cvar[o]);
        }
        for (int k = 0; k < LL; ++k) {
            // Right child, label row k (loaded per (j,k): keeps arrays in regs)
            float rkw[CC], rkmu[CC], v1r[CC];
#pragma unroll
            for (int cr = 0; cr < CC; ++cr) {
                const size_t o = rOff + k * CC + cr;
                rkw[cr]  = cw[o];
                rkmu[cr] = cmu[o];
                v1r[cr]  = __expf(2.0f * cvar[o]);
            }

            const int base0 = ((i * LL + j) * LL + k) * CC;
            float cs[CC];
#pragma unroll
            for (int c = 0; c < CC; ++c) {
                const int base = base0 + c;
                // ---- left logsumexp over cl ----
                float tmu = t_mu_lc[base];
                float v2  = __expf(2.0f * t_var_lc[base]);
                float xs[CC], mx = NEG_INF;
#pragma unroll
                for (int cl = 0; cl < CC; ++cl) {
                    float add = v1l[cl] + v2;
                    float lga = __logf(add);
                    float d   = ljmu[cl] - tmu;
                    float sc  = -0.5f * (LOG2PI_F + lga + d * d / add);
                    float x   = ljw[cl] + sc;
                    xs[cl] = x;
                    mx = fmaxf(mx, x);
                }
                float s = 0.0f;
#pragma unroll
                for (int cl = 0; cl < CC; ++cl) s += __expf(xs[cl] - mx);
                const float lp = mx + __logf(s);

                // ---- right logsumexp over cr ----
                tmu = t_mu_rc[base];
                v2  = __expf(2.0f * t_var_rc[base]);
                mx  = NEG_INF;
#pragma unroll
                for (int cr = 0; cr < CC; ++cr) {
                    float add = v1r[cr] + v2;
                    float lga = __logf(add);
                    float d   = rkmu[cr] - tmu;
                    float sc  = -0.5f * (LOG2PI_F + lga + d * d / add);
                    float x   = rkw[cr] + sc;
                    xs[cr] = x;
                    mx = fmaxf(mx, x);
                }
                s = 0.0f;
#pragma unroll
                for (int cr = 0; cr < CC; ++cr) s += __expf(xs[cr] - mx);
                const float rp = mx + __logf(s);

                cs[c] = lp + rp + t_w[base];
            }

            // ---- parent gaussian + stable top-4 insert ----
#pragma unroll
            for (int cp = 0; cp < CC; ++cp) {
#pragma unroll
                for (int c = 0; c < CC; ++c) {
                    const int base = base0 + c;
                    float v2  = __expf(2.0f * t_var_p[base]);
                    float add = v1p[cp] + v2;
                    float lga = __logf(add);
                    float d   = pmu[cp] - t_mu_p[base];
                    float sc  = -0.5f * (LOG2PI_F + lga + d * d / add);
                    float score = sc + cs[c] + pw[cp];
                    const int idx = ((j * LL + k) * CC + cp) * CC + c;
                    // strict '>' => earlier index wins ties (matches lax.top_k)
                    if (score > best_v[3]) {
                        if (score > best_v[0]) {
                            best_v[3]=best_v[2]; best_i[3]=best_i[2];
                            best_v[2]=best_v[1]; best_i[2]=best_i[1];
                            best_v[1]=best_v[0]; best_i[1]=best_i[0];
                            best_v[0]=score;     best_i[0]=idx;
                        } else if (score > best_v[1]) {
                            best_v[3]=best_v[2]; best_i[3]=best_i[2];
                            best_v[2]=best_v[1]; best_i[2]=best_i[1];
                            best_v[1]=score;     best_i[1]=idx;
                        } else if (score > best_v[2]) {
                            best_v[3]=best_v[2]; best_i[3]=best_i[2];
                            best_v[2]=score;     best_i[2]=idx;
                        } else {
                            best_v[3]=score;     best_i[3]=idx;
                        }
                    }
                }
            }
        }
    }

    // Recompute mu/var for the 4 winners and write results
    const size_t oOff = ((size_t)b * M + m) * NCOL + i * CC;
#pragma unroll
    for (int tq = 0; tq < CC; ++tq) {
        const int idx = best_i[tq];
        const int c   = idx & 3;
        const int cp  = (idx >> 2) & 3;
        const int k   = (idx >> 4) % LL;
        const int j   = idx / (LL * CC * CC);
        const int base = ((i * LL + j) * LL + k) * CC + c;
        float v2  = __expf(2.0f * t_var_p[base]);
        float add = v1p[cp] + v2;
        float lga = __logf(add);
        float mu  = (pmu[cp] * v2 + t_mu_p[base] * v1p[cp]) / add;
        float var = pvr[cp] + t_var_p[base] - 0.5f * lga;
        ow[oOff + tq]  = best_v[tq];
        omu[oOff + tq] = mu;
        ovar[oOff + tq] = var;
    }
}

// ---------------------------------------------------------------------------
// Kernel 3: root combination + final logsumexp -> out (B, L)
// ---------------------------------------------------------------------------
__global__ void lveg_root(
    const float* __restrict__ cw, const float* __restrict__ cmu,
    const float* __restrict__ cvar,
    const float* __restrict__ t_rw, const float* __restrict__ t_rmu,
    const float* __restrict__ t_rvar,
    float* __restrict__ out)
{
    const int t = blockIdx.x * blockDim.x + threadIdx.x;
    if (t >= BATCH * LL) return;
    const int l = t % LL;
    const int b = t / LL;

    const size_t off = (size_t)b * NCOL + l * CC;
    float xs[CC * CC], mx = NEG_INF;
    int n = 0;
#pragma unroll
    for (int c = 0; c < CC; ++c) {
        const float w   = cw[off + c];
        const float mu1 = cmu[off + c];
        const float v1  = __expf(2.0f * cvar[off + c]);
#pragma unroll
        for (int c2 = 0; c2 < CC; ++c2) {
            const int rb = l * CC + c2;
            float v2  = __expf(2.0f * t_rvar[rb]);
            float add = v1 + v2;
            float lga = __logf(add);
            float d   = mu1 - t_rmu[rb];
            float sc  = -0.5f * (LOG2PI_F + lga + d * d / add);
            float x   = w + t_rw[rb] + sc;
            xs[n++] = x;
            mx = fmaxf(mx, x);
        }
    }
    float s = 0.0f;
#pragma unroll
    for (int q = 0; q < CC * CC; ++q) s += __expf(xs[q] - mx);
    out[(size_t)b * LL + l] = mx + __logf(s);
}

// ---------------------------------------------------------------------------
// Host launcher
// ---------------------------------------------------------------------------
extern "C" void kernel_launch(void* const* d_in, const int* in_sizes, int n_in,
                              void* d_out, int out_size, void* d_ws, size_t ws_size,
                              hipStream_t stream)
{
    const float* hidden  = (const float*)d_in[0];
    const float* W_w     = (const float*)d_in[1];
    const float* b_w     = (const float*)d_in[2];
    const float* W_mu    = (const float*)d_in[3];
    const float* b_mu    = (const float*)d_in[4];
    const float* W_var   = (const float*)d_in[5];
    const float* b_var   = (const float*)d_in[6];
    const float* t_w     = (const float*)d_in[7];
    const float* t_mu_lc = (const float*)d_in[8];
    const float* t_var_lc= (const float*)d_in[9];
    const float* t_mu_rc = (const float*)d_in[10];
    const float* t_var_rc= (const float*)d_in[11];
    const float* t_mu_p  = (const float*)d_in[12];
    const float* t_var_p = (const float*)d_in[13];
    const float* t_rw    = (const float*)d_in[14];
    const float* t_rmu   = (const float*)d_in[15];
    const float* t_rvar  = (const float*)d_in[16];

    float* ws = (float*)d_ws;
    const size_t PROJ = (size_t)NROWS * NCOL;        // 652800 floats each
    float* sw   = ws;
    float* smu  = ws + PROJ;
    float* svar = ws + 2 * PROJ;

    const size_t S = (size_t)BATCH * 64 * NCOL;      // 163840 floats (max level)
    float* pb = ws + 3 * PROJ;                       // two ping-pong triples
    float* bufW[2]  = { pb,          pb + 3 * S };
    float* bufMu[2] = { pb + S,      pb + 4 * S };
    float* bufVr[2] = { pb + 2 * S,  pb + 5 * S };

    // Phase 1: projection GEMM (WMMA f32 16x16x4), 32640/16 = 2040 tiles
    lveg_proj_gemm<<<NROWS / 16, 128, 0, stream>>>(
        hidden, W_w, b_w, W_mu, b_mu, W_var, b_var, sw, smu, svar);

    // Phase 2: bottom-up tree contraction
    const float* inW = sw;
    const float* inMu = smu;
    const float* inVr = svar;
    int rowStride = NNODES;   // child buffer node count per batch row
    int base      = 127;      // leaf_start for first level
    int cur = 0;
    for (int d = 6; d >= 0; --d) {
        const int M     = 1 << d;
        const int start = M - 1;
        const int total = BATCH * M * LL;
        const int blocks = (total + 255) / 256;
        lveg_tree_level<<<blocks, 256, 0, stream>>>(
            inW, inMu, inVr, rowStride, base,
            sw, smu, svar, start, M,
            t_w, t_mu_lc, t_var_lc, t_mu_rc, t_var_rc, t_mu_p, t_var_p,
            bufW[cur], bufMu[cur], bufVr[cur]);
        inW = bufW[cur]; inMu = bufMu[cur]; inVr = bufVr[cur];
        rowStride = M;
        base = 0;
        cur ^= 1;
    }

    // Phase 3: root
    lveg_root<<<(BATCH * LL + 127) / 128, 128, 0, stream>>>(
        inW, inMu, inVr, t_rw, t_rmu, t_rvar, (float*)d_out);
}